// LFA_10445360464114
// MI455X (gfx1250) — compile-verified
//
#include <hip/hip_runtime.h>
#include <hip/hip_bf16.h>

#define BB    2
#define NN    16384
#define KNB   16
#define DD    64
#define DIN   32
#define EPSBN 1e-5f
#define SLOPE 0.2f

typedef float v2f __attribute__((ext_vector_type(2)));
typedef float v8f __attribute__((ext_vector_type(8)));

static __device__ __forceinline__ v8f wmma_f32_k4(v2f a, v2f b, v8f c) {
  // D = A(16x4, f32) * B(4x16, f32) + C(16x16, f32)
  return __builtin_amdgcn_wmma_f32_16x16x4_f32(false, a, false, b, (short)0, c,
                                               false, false);
}

// ---------------------------------------------------------------------------
// Stage A: f = relu(bn1(W1@feat)); q/k/v = W@f + b.  wave = 16 points.
// ---------------------------------------------------------------------------
__global__ __launch_bounds__(256) void lfa_stageA(
    const float* __restrict__ feat, const float* __restrict__ W1,
    const float* __restrict__ bn1,
    const float* __restrict__ Wq, const float* __restrict__ bq,
    const float* __restrict__ Wk, const float* __restrict__ bk,
    const float* __restrict__ Wv, const float* __restrict__ bv,
    float* __restrict__ f_ws, float* __restrict__ q_ws,
    float* __restrict__ k_ws, float* __restrict__ v_ws)
{
  __shared__ float sWbuf[64 * 64];         // staged weight matrix
  __shared__ float ftile[8][64][16];       // per-wave f tile (chan-major)
  __shared__ float sS[64], sT[64];         // bn1 scale/shift

  const int tid  = threadIdx.x;
  const int wave = tid >> 5;
  const int ln   = tid & 31;
  const int col  = ln & 15;
  const int half = ln >> 4;

  for (int i = tid; i < 64 * 32; i += 256) sWbuf[i] = W1[i];
  if (tid < 64) {
    float g = bn1[tid], be = bn1[64 + tid], m = bn1[128 + tid], va = bn1[192 + tid];
    float s = g * rsqrtf(va + EPSBN);
    sS[tid] = s; sT[tid] = be - m * s;
  }
  __syncthreads();

  const int p0 = blockIdx.x * 128 + wave * 16;   // global point base
  const int b  = p0 / NN;
  const int nn = p0 - b * NN;
  const float* featb = feat + (size_t)b * DIN * NN;

  // B fragments of feature (32 rows x 16 cols)
  v2f bf[8];
  #pragma unroll
  for (int kk = 0; kk < 8; ++kk) {
    int c0 = kk * 4 + 2 * half;
    bf[kk].x = featb[(size_t)c0 * NN + nn + col];
    bf[kk].y = featb[(size_t)(c0 + 1) * NN + nn + col];
  }

  #pragma unroll
  for (int mt = 0; mt < 4; ++mt) {
    v8f acc = {0.f, 0.f, 0.f, 0.f, 0.f, 0.f, 0.f, 0.f};
    const int row = mt * 16 + col;
    #pragma unroll
    for (int kk = 0; kk < 8; ++kk) {
      v2f a;
      a.x = sWbuf[row * 32 + kk * 4 + 2 * half];
      a.y = sWbuf[row * 32 + kk * 4 + 2 * half + 1];
      acc = wmma_f32_k4(a, bf[kk], acc);
    }
    #pragma unroll
    for (int v = 0; v < 8; ++v) {
      int chan = mt * 16 + v + 8 * half;
      float fv = sS[chan] * acc[v] + sT[chan];
      fv = fv > 0.f ? fv : 0.f;                  // ReLU
      ftile[wave][chan][col] = fv;
      f_ws[(size_t)(p0 + col) * DD + chan] = fv;
    }
  }
  __syncthreads();

  // B fragments of f (64 rows x 16 cols) from LDS, reused for q/k/v
  v2f bfr[16];
  #pragma unroll
  for (int kk = 0; kk < 16; ++kk) {
    bfr[kk].x = ftile[wave][kk * 4 + 2 * half][col];
    bfr[kk].y = ftile[wave][kk * 4 + 2 * half + 1][col];
  }

  const float* Ws[3]   = {Wq, Wk, Wv};
  const float* bs[3]   = {bq, bk, bv};
  float*       outs[3] = {q_ws, k_ws, v_ws};

  #pragma unroll
  for (int w = 0; w < 3; ++w) {
    __syncthreads();
    for (int i = tid; i < 64 * 64; i += 256) sWbuf[i] = Ws[w][i];
    __syncthreads();
    #pragma unroll
    for (int mt = 0; mt < 4; ++mt) {
      v8f acc = {0.f, 0.f, 0.f, 0.f, 0.f, 0.f, 0.f, 0.f};
      const int row = mt * 16 + col;
      #pragma unroll
      for (int kk = 0; kk < 16; ++kk) {
        v2f a;
        a.x = sWbuf[row * 64 + kk * 4 + 2 * half];
        a.y = sWbuf[row * 64 + kk * 4 + 2 * half + 1];
        acc = wmma_f32_k4(a, bfr[kk], acc);
      }
      #pragma unroll
      for (int v = 0; v < 8; ++v) {
        int chan = mt * 16 + v + 8 * half;
        outs[w][(size_t)(p0 + col) * DD + chan] = acc[v] + bs[w][chan];
      }
    }
  }
}

// ---------------------------------------------------------------------------
// Stage B: neighbor gather + p_r + attention weights (WMMA on Ww1) + softmax
// + grouped weighted sum.  wave = 1 point (its 16 neighbors = 16 cols).
// ---------------------------------------------------------------------------
__global__ __launch_bounds__(128) void lfa_stageB(
    const float* __restrict__ xyz, const int* __restrict__ nidx,
    const float* __restrict__ Wp1, const float* __restrict__ bnp1,
    const float* __restrict__ Wp2, const float* __restrict__ bp2,
    const float* __restrict__ bnw0,
    const float* __restrict__ Ww1, const float* __restrict__ bnw1,
    const float* __restrict__ Ww2, const float* __restrict__ bw2,
    const float* __restrict__ q_ws, const float* __restrict__ k_ws,
    const float* __restrict__ v_ws, float* __restrict__ x_ws)
{
  __shared__ float sWw1[16 * 64];            // Ww1 padded to 16 rows
  __shared__ float sWw2[64];
  __shared__ float sWp1[9], sSp1[3], sTp1[3];
  __shared__ float sWp2[192], sbp2[64];
  __shared__ float sSw0[64], sTw0[64];
  __shared__ float sSw1[8], sTw1[8], sbw2[8];
  __shared__ float w0buf[4][64][16];
  __shared__ float xvbuf[4][64][16];
  __shared__ float hbuf[4][3][16];
  __shared__ float qbuf[4][64];
  __shared__ int   idxbuf[4][16];
  __shared__ float w1buf[4][8][16];
  __shared__ float smbuf[4][8][16];

  const int tid  = threadIdx.x;
  const int wave = tid >> 5;
  const int ln   = tid & 31;
  const int col  = ln & 15;
  const int half = ln >> 4;

  #pragma unroll
  for (int i = 0; i < 8; ++i) {
    int idx = tid + 128 * i;                  // 1024 entries
    int row = idx >> 6, d = idx & 63;
    sWw1[idx] = (row < 8) ? Ww1[row * 64 + d] : 0.f;
  }
  if (tid < 64) {
    sWw2[tid] = Ww2[tid & 63];
    sbp2[tid] = bp2[tid];
    float g = bnw0[tid], be = bnw0[64 + tid], m = bnw0[128 + tid], va = bnw0[192 + tid];
    float s = g * rsqrtf(va + EPSBN);
    sSw0[tid] = s; sTw0[tid] = be - m * s;
  }
  if (tid < 9) sWp1[tid] = Wp1[tid];
  if (tid < 3) {
    float g = bnp1[tid], be = bnp1[3 + tid], m = bnp1[6 + tid], va = bnp1[9 + tid];
    float s = g * rsqrtf(va + EPSBN);
    sSp1[tid] = s; sTp1[tid] = be - m * s;
  }
  if (tid < 192) sWp2[tid] = Wp2[tid];
  if (tid < 8) {
    float g = bnw1[tid], be = bnw1[8 + tid], m = bnw1[16 + tid], va = bnw1[24 + tid];
    float s = g * rsqrtf(va + EPSBN);
    sSw1[tid] = s; sTw1[tid] = be - m * s;
    sbw2[tid] = bw2[tid];
  }
  __syncthreads();

  const int p = blockIdx.x * 4 + wave;        // global point
  const int b = p / NN;

  if (ln < 16) {
    int idx = nidx[(size_t)p * KNB + ln];
    idxbuf[wave][ln] = idx;
    const float* xp = xyz + ((size_t)b * NN + idx) * 3;
    float x0 = xp[0], x1 = xp[1], x2 = xp[2];
    #pragma unroll
    for (int c = 0; c < 3; ++c) {
      float hr = sWp1[c * 3] * x0 + sWp1[c * 3 + 1] * x1 + sWp1[c * 3 + 2] * x2;
      float h  = sSp1[c] * hr + sTp1[c];
      hbuf[wave][c][ln] = h > 0.f ? h : 0.f;
    }
  }
  qbuf[wave][ln]      = q_ws[(size_t)p * DD + ln];
  qbuf[wave][ln + 32] = q_ws[(size_t)p * DD + ln + 32];
  __syncthreads();

  const int kidx = idxbuf[wave][col];
  const size_t kb = ((size_t)b * NN + kidx) * DD;
  const float h0 = hbuf[wave][0][col];
  const float h1 = hbuf[wave][1][col];
  const float h2 = hbuf[wave][2][col];

  #pragma unroll
  for (int i = 0; i < 32; ++i) {
    int d = 2 * i + half;
    float kg = k_ws[kb + d];
    float vg = v_ws[kb + d];
    float pr = sWp2[d * 3] * h0 + sWp2[d * 3 + 1] * h1 + sWp2[d * 3 + 2] * h2 + sbp2[d];
    float r  = kg - qbuf[wave][d] + pr;
    float t  = sSw0[d] * r + sTw0[d];
    w0buf[wave][d][col] = t >= 0.f ? t : SLOPE * t;   // LeakyReLU
    xvbuf[wave][d][col] = vg + pr;
  }
  __syncthreads();

  // w1 = Ww1_pad(16x64) @ w0(64x16) via 16 chained WMMAs
  v8f acc = {0.f, 0.f, 0.f, 0.f, 0.f, 0.f, 0.f, 0.f};
  #pragma unroll
  for (int kk = 0; kk < 16; ++kk) {
    v2f a, bfr;
    a.x   = sWw1[col * 64 + kk * 4 + 2 * half];
    a.y   = sWw1[col * 64 + kk * 4 + 2 * half + 1];
    bfr.x = w0buf[wave][kk * 4 + 2 * half][col];
    bfr.y = w0buf[wave][kk * 4 + 2 * half + 1][col];
    acc = wmma_f32_k4(a, bfr, acc);
  }
  if (half == 0) {                            // rows 0..7 are the valid outputs
    #pragma unroll
    for (int v = 0; v < 8; ++v) {
      float t = sSw1[v] * acc[v] + sTw1[v];
      w1buf[wave][v][col] = t > 0.f ? t : 0.f;
    }
  }
  __syncthreads();

  if (ln < 8) {                               // 8x8 conv + softmax over K
    float wv[16];
    float mx = -1e30f;
    #pragma unroll
    for (int k2 = 0; k2 < 16; ++k2) {
      float s = sbw2[ln];
      #pragma unroll
      for (int j = 0; j < 8; ++j) s += sWw2[ln * 8 + j] * w1buf[wave][j][k2];
      wv[k2] = s;
      mx = fmaxf(mx, s);
    }
    float sum = 0.f;
    #pragma unroll
    for (int k2 = 0; k2 < 16; ++k2) { wv[k2] = __expf(wv[k2] - mx); sum += wv[k2]; }
    float inv = 1.f / sum;
    #pragma unroll
    for (int k2 = 0; k2 < 16; ++k2) smbuf[wave][ln][k2] = wv[k2] * inv;
  }
  __syncthreads();

  #pragma unroll
  for (int t2 = 0; t2 < 2; ++t2) {
    int d = ln + 32 * t2;
    float s = 0.f;
    #pragma unroll
    for (int k2 = 0; k2 < 16; ++k2)
      s += xvbuf[wave][d][k2] * smbuf[wave][d & 7][k2];
    x_ws[(size_t)p * DD + d] = s;
  }
}

// ---------------------------------------------------------------------------
// Stage C: y = lrelu(bn_mid(x)); out = lrelu(f + bn2(W2@y)).  wave = 16 points.
// ---------------------------------------------------------------------------
__global__ __launch_bounds__(256) void lfa_stageC(
    const float* __restrict__ x_ws, const float* __restrict__ f_ws,
    const float* __restrict__ bn_mid, const float* __restrict__ W2,
    const float* __restrict__ bn2, float* __restrict__ out)
{
  __shared__ float sW2[64 * 64];
  __shared__ float ytile[8][64][16];
  __shared__ float sSm[64], sTm[64], sS2[64], sT2[64];

  const int tid  = threadIdx.x;
  const int wave = tid >> 5;
  const int ln   = tid & 31;
  const int col  = ln & 15;
  const int half = ln >> 4;

  for (int i = tid; i < 64 * 64; i += 256) sW2[i] = W2[i];
  if (tid < 64) {
    float g = bn_mid[tid], be = bn_mid[64 + tid], m = bn_mid[128 + tid], va = bn_mid[192 + tid];
    float s = g * rsqrtf(va + EPSBN);
    sSm[tid] = s; sTm[tid] = be - m * s;
    g = bn2[tid]; be = bn2[64 + tid]; m = bn2[128 + tid]; va = bn2[192 + tid];
    s = g * rsqrtf(va + EPSBN);
    sS2[tid] = s; sT2[tid] = be - m * s;
  }
  __syncthreads();

  const int p0 = blockIdx.x * 128 + wave * 16;
  const int b  = p0 / NN;
  const int nn = p0 - b * NN;

  #pragma unroll
  for (int i = 0; i < 32; ++i) {
    int d = 2 * i + half;
    float xv = x_ws[(size_t)(p0 + col) * DD + d];
    float t  = sSm[d] * xv + sTm[d];
    ytile[wave][d][col] = t >= 0.f ? t : SLOPE * t;   // LeakyReLU
  }
  __syncthreads();

  v2f bfr[16];
  #pragma unroll
  for (int kk = 0; kk < 16; ++kk) {
    bfr[kk].x = ytile[wave][kk * 4 + 2 * half][col];
    bfr[kk].y = ytile[wave][kk * 4 + 2 * half + 1][col];
  }

  #pragma unroll
  for (int mt = 0; mt < 4; ++mt) {
    v8f acc = {0.f, 0.f, 0.f, 0.f, 0.f, 0.f, 0.f, 0.f};
    const int row = mt * 16 + col;
    #pragma unroll
    for (int kk = 0; kk < 16; ++kk) {
      v2f a;
      a.x = sW2[row * 64 + kk * 4 + 2 * half];
      a.y = sW2[row * 64 + kk * 4 + 2 * half + 1];
      acc = wmma_f32_k4(a, bfr[kk], acc);
    }
    #pragma unroll
    for (int v = 0; v < 8; ++v) {
      int chan = mt * 16 + v + 8 * half;
      float z  = sS2[chan] * acc[v] + sT2[chan];
      float fv = f_ws[(size_t)(p0 + col) * DD + chan];
      float o  = fv + z;
      out[(size_t)b * DD * NN + (size_t)chan * NN + nn + col] =
          o >= 0.f ? o : SLOPE * o;
    }
  }
}

// ---------------------------------------------------------------------------
extern "C" void kernel_launch(void* const* d_in, const int* in_sizes, int n_in,
                              void* d_out, int out_size, void* d_ws, size_t ws_size,
                              hipStream_t stream) {
  const float* feature = (const float*)d_in[0];
  const float* xyz     = (const float*)d_in[1];
  const float* W1      = (const float*)d_in[2];
  const float* bn1     = (const float*)d_in[3];
  const float* Wq      = (const float*)d_in[4];
  const float* bq      = (const float*)d_in[5];
  const float* Wk      = (const float*)d_in[6];
  const float* bk      = (const float*)d_in[7];
  const float* Wv      = (const float*)d_in[8];
  const float* bv      = (const float*)d_in[9];
  const float* Wp1     = (const float*)d_in[10];
  const float* bnp1    = (const float*)d_in[11];
  const float* Wp2     = (const float*)d_in[12];
  const float* bp2     = (const float*)d_in[13];
  const float* bnw0    = (const float*)d_in[14];
  const float* Ww1     = (const float*)d_in[15];
  const float* bnw1    = (const float*)d_in[16];
  const float* Ww2     = (const float*)d_in[17];
  const float* bw2     = (const float*)d_in[18];
  const float* bn_mid  = (const float*)d_in[19];
  const float* W2      = (const float*)d_in[20];
  const float* bn2     = (const float*)d_in[21];
  const int*   nidx    = (const int*)d_in[22];
  float* out = (float*)d_out;

  float* ws = (float*)d_ws;
  const size_t S = (size_t)BB * NN * DD;      // 2,097,152 floats per buffer
  float* f_ws = ws;
  float* q_ws = ws + S;
  float* k_ws = ws + 2 * S;
  float* v_ws = ws + 3 * S;
  float* x_ws = ws + 4 * S;

  const int total_pts = BB * NN;              // 32768

  lfa_stageA<<<total_pts / 128, 256, 0, stream>>>(
      feature, W1, bn1, Wq, bq, Wk, bk, Wv, bv, f_ws, q_ws, k_ws, v_ws);

  lfa_stageB<<<total_pts / 4, 128, 0, stream>>>(
      xyz, nidx, Wp1, bnp1, Wp2, bp2, bnw0, Ww1, bnw1, Ww2, bw2,
      q_ws, k_ws, v_ws, x_ws);

  lfa_stageC<<<total_pts / 128, 256, 0, stream>>>(
      x_ws, f_ws, bn_mid, W2, bn2, out);
}